// PoolingByAgreement_44023414784336
// MI455X (gfx1250) — compile-verified
//
#include <hip/hip_runtime.h>
#include <math.h>

typedef float v2f __attribute__((ext_vector_type(2)));
typedef float v8f __attribute__((ext_vector_type(8)));

// Problem dims (fixed by reference)
constexpr int B  = 64;
constexpr int O  = 32;
constexpr int I  = 1024;
constexpr int CI = 16;
constexpr int CO = 16;
constexpr int ITER = 3;

// batch stride in u_hat[B][O][I][CO], in floats
constexpr size_t BSTRIDE = (size_t)O * I * CO;   // 512K floats = 2MB

// ---------------------------------------------------------------------------
// Kernel 1: u_hat[b,o,i,d] = sum_c data[b,i,c] * W[o,i,c,d]
// Per (o,i): [64x16] @ [16x16] GEMM -> 4 M-tiles x 4 K-steps of
// V_WMMA_F32_16X16X4_F32. One wave owns one i and 8 consecutive o's so the
// A fragments (data) are loaded once and reused across o.
// Memory-bound: 33.5MB W read + 134MB u_hat write (stays resident in L2).
// ---------------------------------------------------------------------------
__global__ __launch_bounds__(256) void uhat_wmma_kernel(
    const float* __restrict__ data,   // [B][I][CI]
    const float* __restrict__ W,      // [O][I][CI][CO]
    float* __restrict__ uhat)         // [B][O][I][CO]
{
    const int wave = threadIdx.x >> 5;
    const int lane = threadIdx.x & 31;
    const int gw   = blockIdx.x * 8 + wave;   // 0..4095
    const int i    = gw >> 2;                 // 0..1023
    const int oc   = gw & 3;                  // o-chunk: o = oc*8 .. oc*8+7
    const int half = lane >> 4;               // 0: lanes 0-15, 1: lanes 16-31
    const int lr   = lane & 15;

    // A fragments: 16x4 f32 tile per (mt, ks).
    // Lane layout (ISA 7.12.2): lanes 0-15 hold M=lr, K={4ks+0, 4ks+1};
    // lanes 16-31 hold M=lr, K={4ks+2, 4ks+3}.  M maps to batch mt*16+lr.
    v2f a[4][4];
#pragma unroll
    for (int mt = 0; mt < 4; ++mt)
#pragma unroll
        for (int ks = 0; ks < 4; ++ks) {
            const int bb = mt * 16 + lr;
            const float* p = data + ((size_t)bb * I + i) * CI + ks * 4 + half * 2;
            a[mt][ks] = *(const v2f*)p;
        }

    // Per-o output base for this lane: rows bo = half*8 (+r, +16*mt) at d=lr.
    // Consecutive batches are BSTRIDE floats (2MB) apart -> constant offsets.
    float* pbase0 = uhat + (((size_t)(half * 8) * O) * I + i) * CO + lr;

    for (int oo = 0; oo < 8; ++oo) {
        const int o = oc * 8 + oo;

        // B fragments: 4x16 f32 tile per ks, row-striped like C/D:
        // v0: lanes 0-15 row c=4ks+0 / lanes 16-31 row c=4ks+2 (N = lr)
        // v1: rows 4ks+1 / 4ks+3.
        v2f bf[4];
#pragma unroll
        for (int ks = 0; ks < 4; ++ks) {
            const float* q = W + (((size_t)o * I + i) * CI + ks * 4 + half * 2) * CO + lr;
            v2f t;
            t.x = q[0];
            t.y = q[CO];
            bf[ks] = t;
        }

        v8f acc[4] = {};
#pragma unroll
        for (int ks = 0; ks < 4; ++ks)
#pragma unroll
            for (int mt = 0; mt < 4; ++mt)
                acc[mt] = __builtin_amdgcn_wmma_f32_16x16x4_f32(
                    /*neg_a=*/false, a[mt][ks],
                    /*neg_b=*/false, bf[ks],
                    /*c_mod=*/(short)0, acc[mt],
                    /*reuse_a=*/false, /*reuse_b=*/false);

        // D layout: VGPR r -> M=r (lanes 0-15) / M=r+8 (lanes 16-31), N=lr.
        // One base per (mt,o); row step is a compile-time constant stride so
        // the compiler can fold it into store offsets / single 64-bit adds.
        float* po = pbase0 + (size_t)o * I * CO;
#pragma unroll
        for (int mt = 0; mt < 4; ++mt) {
            float* p = po + (size_t)(mt * 16) * BSTRIDE;
#pragma unroll
            for (int r = 0; r < 8; ++r)
                p[(size_t)r * BSTRIDE] = acc[mt][r];
        }
    }
}

// ---------------------------------------------------------------------------
// Kernel 2: fused dynamic routing. One workgroup per batch element.
// b_log (32x1024 f32 = 128KB) lives entirely in LDS. Wave w owns o=w.
// u_hat is streamed from L2 (it fits: 134MB < 192MB).
// ---------------------------------------------------------------------------
__global__ __launch_bounds__(1024) void routing_kernel(
    const float* __restrict__ uhat,   // [B][O][I][CO]
    const float* __restrict__ bias,   // [O]
    const float* __restrict__ beta,   // [O]
    const float* __restrict__ alpha,  // [O]
    float* __restrict__ out)          // [B][O][CO]
{
    __shared__ float blog[O * I];     // 128 KB  routing logits
    __shared__ float mbuf[I];         // softmax max per i
    __shared__ float zbuf[I];         // softmax 1/Z per i
    __shared__ float sbuf[O * CO];    // s vectors
    __shared__ float vbuf[O * CO];    // v vectors

    const int b    = blockIdx.x;
    const int tid  = threadIdx.x;
    const int w    = tid >> 5;        // wave id == output capsule o
    const int lane = tid & 31;

    // init: b_log = 0 -> softmax is uniform: m=0, 1/Z = 1/32
    for (int idx = tid; idx < O * I; idx += 1024) blog[idx] = 0.0f;
    mbuf[tid & (I - 1)] = 0.0f;
    zbuf[tid & (I - 1)] = 1.0f / 32.0f;
    __syncthreads();

    const float* ub = uhat + ((size_t)b * O + w) * I * CO;  // this wave's o-slice

    for (int t = 0; t < ITER; ++t) {
        // ---- pass A: s[o,:] = sum_i c[o,i] * u_hat[o,i,:] ----
        float acc[CO];
#pragma unroll
        for (int d = 0; d < CO; ++d) acc[d] = 0.0f;

        for (int i = lane; i < I; i += 32) {
            const float c = __expf(blog[w * I + i] - mbuf[i]) * zbuf[i];
            const float4* up = (const float4*)(ub + (size_t)i * CO);
            const float4 u0 = up[0], u1 = up[1], u2 = up[2], u3 = up[3];
            acc[0]  += c * u0.x;  acc[1]  += c * u0.y;  acc[2]  += c * u0.z;  acc[3]  += c * u0.w;
            acc[4]  += c * u1.x;  acc[5]  += c * u1.y;  acc[6]  += c * u1.z;  acc[7]  += c * u1.w;
            acc[8]  += c * u2.x;  acc[9]  += c * u2.y;  acc[10] += c * u2.z;  acc[11] += c * u2.w;
            acc[12] += c * u3.x;  acc[13] += c * u3.y;  acc[14] += c * u3.z;  acc[15] += c * u3.w;
        }
        // wave32 butterfly reduce
#pragma unroll
        for (int off = 16; off > 0; off >>= 1)
#pragma unroll
            for (int d = 0; d < CO; ++d)
                acc[d] += __shfl_xor(acc[d], off, 32);
        if (lane < CO) sbuf[w * CO + lane] = acc[lane];
        __syncthreads();

        // ---- squash: v = (|s|^2/(1+|s|^2)) * s / sqrt(|s|^2+eps) ----
        if (tid < O * CO) {
            const int o = tid >> 4;
            const int d = tid & 15;
            const float bo = bias[o];
            float sq = 0.0f;
#pragma unroll
            for (int dd = 0; dd < CO; ++dd) {
                const float s = sbuf[o * CO + dd] + bo;
                sq += s * s;
            }
            const float s = sbuf[o * CO + d] + bo;
            vbuf[tid] = (sq / (1.0f + sq)) * s * rsqrtf(sq + 1e-8f);
        }
        __syncthreads();

        if (t < ITER - 1) {
            // ---- pass B: b_log[o,i] += dot(v[o,:], u_hat[o,i,:]) ----
            float v[CO];
#pragma unroll
            for (int d = 0; d < CO; ++d) v[d] = vbuf[w * CO + d];  // LDS broadcast

            for (int i = lane; i < I; i += 32) {
                const float4* up = (const float4*)(ub + (size_t)i * CO);
                const float4 u0 = up[0], u1 = up[1], u2 = up[2], u3 = up[3];
                float dot =
                    v[0]  * u0.x + v[1]  * u0.y + v[2]  * u0.z + v[3]  * u0.w +
                    v[4]  * u1.x + v[5]  * u1.y + v[6]  * u1.z + v[7]  * u1.w +
                    v[8]  * u2.x + v[9]  * u2.y + v[10] * u2.z + v[11] * u2.w +
                    v[12] * u3.x + v[13] * u3.y + v[14] * u3.z + v[15] * u3.w;
                blog[w * I + i] += dot;
            }
            __syncthreads();

            // ---- recompute softmax stats over O per input capsule i ----
            {
                const int i = tid;  // blockDim == I
                float mx = -INFINITY;
                for (int o = 0; o < O; ++o) mx = fmaxf(mx, blog[o * I + i]);
                float sum = 0.0f;
                for (int o = 0; o < O; ++o) sum += __expf(blog[o * I + i] - mx);
                mbuf[i] = mx;
                zbuf[i] = 1.0f / sum;
            }
            __syncthreads();
        }
    }

    // ---- activation gate + output ----
    if (tid < O * CO) {
        const int o = tid >> 4;
        float sq = 1e-8f;
#pragma unroll
        for (int dd = 0; dd < CO; ++dd) {
            const float vv = vbuf[o * CO + dd];
            sq += vv * vv;
        }
        const float norm = sqrtf(sq);
        const float act  = 1.0f / (1.0f + __expf(-(alpha[o] * norm + beta[o])));
        out[(size_t)b * O * CO + tid] = vbuf[tid] * act;
    }
}

// ---------------------------------------------------------------------------
extern "C" void kernel_launch(void* const* d_in, const int* in_sizes, int n_in,
                              void* d_out, int out_size, void* d_ws, size_t ws_size,
                              hipStream_t stream) {
    (void)in_sizes; (void)n_in; (void)out_size; (void)ws_size;
    const float* data  = (const float*)d_in[0];  // [B,I,CI]
    const float* W     = (const float*)d_in[1];  // [O,I,CI,CO]
    const float* bias  = (const float*)d_in[2];  // [O]
    const float* beta  = (const float*)d_in[3];  // [O]
    const float* alpha = (const float*)d_in[4];  // [O]
    float* out  = (float*)d_out;                 // [B,O,CO]
    float* uhat = (float*)d_ws;                  // needs B*O*I*CO*4 = 128 MiB

    // 4096 waves: wave = (i, o-chunk of 8); 8 waves per 256-thread block.
    uhat_wmma_kernel<<<dim3((I * 4) / 8), dim3(256), 0, stream>>>(data, W, uhat);
    // one workgroup per batch element, 32 waves (one per output capsule)
    routing_kernel<<<dim3(B), dim3(1024), 0, stream>>>(uhat, bias, beta, alpha, out);
}